// PixelContrastMorphologyLoss_24885040513178
// MI455X (gfx1250) — compile-verified
//
#include <hip/hip_runtime.h>
#include <hip/hip_bf16.h>
#include <math.h>

typedef __attribute__((ext_vector_type(2))) float v2f;
typedef __attribute__((ext_vector_type(8))) float v8f;

#define HW        16384   // 128*128
#define CCH       256     // channels
#define NVIEW     100
#define NROWS     200     // 2 classes * 100 views
#define NPAD      208     // 13 tiles of 16
#define NTILE     13
#define SSTRIDE   260     // LDS row stride in floats (bank-conflict-free)
#define INV_TEMP  10.0f
#define LOSS_SCL  (0.1f / 0.07f)

extern __shared__ char dyn_smem[];

// ---------------------------------------------------------------------------
// Kernel 1: per (image, class) top-100 selection by sel_noise among mask==cls
// grid = 32 blocks (img*2 + clsIdx), 256 threads. Dyn LDS = HW floats (64KB).
// ---------------------------------------------------------------------------
__global__ __launch_bounds__(256)
void pcml_select(const int* __restrict__ mask_within,
                 const float* __restrict__ sel_noise,
                 int* __restrict__ sel_idx) {
    float* sc = (float*)dyn_smem;                 // HW floats
    __shared__ float rmax[256];
    __shared__ int   ridx[256];

    const int tid = threadIdx.x;
    const int img = blockIdx.x >> 1;
    const int cls = (blockIdx.x & 1) + 1;

    for (int i = tid; i < HW; i += 256) {
        int   m = mask_within[(size_t)img * HW + i];
        float s = sel_noise[(size_t)img * HW + i];
        sc[i] = (m == cls) ? s : -INFINITY;
    }
    __syncthreads();

    for (int v = 0; v < NVIEW; ++v) {
        float best = -INFINITY;
        int   bi   = 0;
        const int base = tid * (HW / 256);
        #pragma unroll 4
        for (int i = 0; i < HW / 256; ++i) {
            float s = sc[base + i];
            if (s > best) { best = s; bi = base + i; }
        }
        rmax[tid] = best; ridx[tid] = bi;
        __syncthreads();
        for (int off = 128; off >= 1; off >>= 1) {
            if (tid < off) {
                if (rmax[tid + off] > rmax[tid]) {
                    rmax[tid] = rmax[tid + off];
                    ridx[tid] = ridx[tid + off];
                }
            }
            __syncthreads();
        }
        if (tid == 0) {
            sel_idx[blockIdx.x * NVIEW + v] = ridx[0];
            sc[ridx[0]] = -INFINITY;              // remove winner
        }
        __syncthreads();
    }
}

// ---------------------------------------------------------------------------
// Kernel 2: gather selected feature rows into LDS, compute Gram via f32 WMMA,
// store logits*(1/TEMP) to workspace. grid = 16 blocks, 256 thr (8 waves).
// Dyn LDS = NPAD*SSTRIDE floats = 216320 bytes (fits CDNA5 320KB WGP LDS).
// ---------------------------------------------------------------------------
__global__ __launch_bounds__(256)
void pcml_gram(const float* __restrict__ feats,
               const int* __restrict__ sel_idx,
               float* __restrict__ logits) {
    float* S = (float*)dyn_smem;                  // [NPAD][SSTRIDE]
    const int tid  = threadIdx.x;
    const int img  = blockIdx.x;

    // Gather: row r = view-major interleave (class = r&1, view = r>>1),
    // matching the reference's torch.cat(torch.unbind(...,1),0) ordering.
    for (int e = tid; e < NPAD * CCH; e += 256) {
        const int r = e >> 8;          // row
        const int c = e & 255;         // channel
        float v = 0.0f;
        if (r < NROWS) {
            const int cls  = r & 1;
            const int view = r >> 1;
            const int p = sel_idx[(img * 2 + cls) * NVIEW + view];
            v = feats[((size_t)img * CCH + c) * HW + p];
        }
        S[r * SSTRIDE + c] = v;
    }
    __syncthreads();

    const int wave = tid >> 5;
    const int lane = tid & 31;
    const int la   = lane & 15;
    const int hi   = lane >> 4;        // lanes 16-31 carry K+2,K+3 (A 16x4 f32 layout)

    for (int t = wave; t < NTILE * NTILE; t += 8) {
        const int mt = t / NTILE;
        const int nt = t % NTILE;
        const int rowA = (mt * 16 + la) * SSTRIDE;
        const int rowB = (nt * 16 + la) * SSTRIDE;
        const int koff = hi * 2;

        v8f acc = {0.f, 0.f, 0.f, 0.f, 0.f, 0.f, 0.f, 0.f};
        #pragma unroll 8
        for (int kb = 0; kb < CCH / 4; ++kb) {
            const int k = (kb << 2) + koff;
            v2f a, b;
            a.x = S[rowA + k]; a.y = S[rowA + k + 1];
            b.x = S[rowB + k]; b.y = S[rowB + k + 1];
            // D = A*B + C ; Gram matrix: B fragment == A fragment of tile nt
            acc = __builtin_amdgcn_wmma_f32_16x16x4_f32(
                false, a, false, b, (short)0, acc, false, false);
        }

        // C/D layout: VGPR i -> M = i + (lane>=16 ? 8 : 0), N = lane&15
        const int n  = nt * 16 + la;
        const int mb = mt * 16 + hi * 8;
        size_t base = ((size_t)img * NPAD + mb) * NPAD + n;
        #pragma unroll
        for (int i = 0; i < 8; ++i)
            logits[base + (size_t)i * NPAD] = acc[i] * INV_TEMP;
    }
}

// ---------------------------------------------------------------------------
// Kernel 3: per-row contrastive loss; block-reduce to per-image sum.
// grid = 16 blocks, 256 threads (thread r handles row r, r < 200).
// ---------------------------------------------------------------------------
__global__ __launch_bounds__(256)
void pcml_rowloss(const float* __restrict__ logits,
                  float* __restrict__ img_loss) {
    __shared__ float red[256];
    const int tid = threadIdx.x;
    const int img = blockIdx.x;

    float rowLoss = 0.0f;
    if (tid < NROWS) {
        const int r = tid;
        const size_t base = ((size_t)img * NPAD + r) * NPAD;

        float m = -INFINITY;
        for (int j = 0; j < NROWS; ++j)
            m = fmaxf(m, logits[base + j]);

        float negSum = 0.0f;
        for (int j = 0; j < NROWS; ++j) {
            const float l = logits[base + j] - m;
            if ((j & 1) != (r & 1)) negSum += expf(l);
        }

        float posAcc = 0.0f;
        for (int j = 0; j < NROWS; ++j) {
            if (((j & 1) == (r & 1)) && (j != r)) {
                const float l = logits[base + j] - m;
                posAcc += l - logf(expf(l) + negSum);
            }
        }
        rowLoss = -LOSS_SCL * (posAcc / (float)(NVIEW - 1));
    }

    red[tid] = rowLoss;
    __syncthreads();
    for (int off = 128; off >= 1; off >>= 1) {
        if (tid < off) red[tid] += red[tid + off];
        __syncthreads();
    }
    if (tid == 0) img_loss[img] = red[0];
}

// ---------------------------------------------------------------------------
// Kernel 4: final scalar = mean over rows (200) and images (16).
// ---------------------------------------------------------------------------
__global__ void pcml_final(const float* __restrict__ img_loss,
                           float* __restrict__ out) {
    if (threadIdx.x == 0) {
        float s = 0.0f;
        for (int i = 0; i < 16; ++i) s += img_loss[i];
        out[0] = s / (float)(16 * NROWS);
    }
}

// ---------------------------------------------------------------------------
extern "C" void kernel_launch(void* const* d_in, const int* in_sizes, int n_in,
                              void* d_out, int out_size, void* d_ws, size_t ws_size,
                              hipStream_t stream) {
    const float* feats       = (const float*)d_in[0];
    const int*   mask_within = (const int*)d_in[1];
    // d_in[2] mask_boundary, d_in[3] predict: unused by the reference math
    const float* sel_noise   = (const float*)d_in[4];

    // workspace layout
    int*   sel_idx  = (int*)d_ws;                                    // 32*100 ints
    float* logits   = (float*)((char*)d_ws + 16384);                 // 16*208*208 f32
    float* img_loss = (float*)((char*)d_ws + 16384 +
                               (size_t)16 * NPAD * NPAD * sizeof(float)); // 16 f32

    // 1) top-100 selection per (image, class)
    pcml_select<<<32, 256, HW * sizeof(float), stream>>>(mask_within, sel_noise, sel_idx);

    // 2) gather + f32 WMMA Gram matrix (logits / TEMP)
    pcml_gram<<<16, 256, NPAD * SSTRIDE * sizeof(float), stream>>>(feats, sel_idx, logits);

    // 3) per-row contrastive loss -> per-image sums
    pcml_rowloss<<<16, 256, 0, stream>>>(logits, img_loss);

    // 4) scalar mean
    pcml_final<<<1, 64, 0, stream>>>(img_loss, (float*)d_out);
}